// GATLayer_2224793059990
// MI455X (gfx1250) — compile-verified
//
#include <hip/hip_runtime.h>

#define IN_DIM 128
#define OUT_DIM 64

typedef float v2f __attribute__((ext_vector_type(2)));
typedef float v8f __attribute__((ext_vector_type(8)));

// ---------------------------------------------------------------------------
// Zero d_out (harness poisons it with 0xAA; scatter-add needs zeros).
// ---------------------------------------------------------------------------
__global__ void gat_zero_kernel(float4* __restrict__ out, int n4) {
  int i = blockIdx.x * blockDim.x + threadIdx.x;
  if (i < n4) {
    out[i] = make_float4(0.0f, 0.0f, 0.0f, 0.0f);
  }
}

// ---------------------------------------------------------------------------
// z = h @ W_fc^T + b_fc  via V_WMMA_F32_16X16X4_F32 (f32 in/out, exact dtype
// match with the reference). One wave computes a 16-row x 64-col slab of z:
// 4 accumulators of 16x16, K swept in steps of 4.
//
// A (16x4 f32) layout: lane%16 = M; k-pair selected by lane/16; vgpr = k LSB.
//   -> lane loads float2 from h[(row0 + lane%16)*128 + k + 2*(lane/16)]
// B (4x16 f32) layout mirrors A with lane%16 = N. B = W_fc^T, so
//   B[k][n] = W_fc[n][k] -> lane loads float2 from
//   W_fc[(col0 + lane%16)*128 + k + 2*(lane/16)]   (contiguous along k)
// C/D (16x16 f32): vgpr r, M = r + 8*(lane/16), N = lane%16.
// ---------------------------------------------------------------------------
__global__ void gat_gemm_wmma_kernel(const float* __restrict__ h,
                                     const float* __restrict__ Wfc,
                                     const float* __restrict__ bfc,
                                     float* __restrict__ z,
                                     int nrows) {
  const int lane  = threadIdx.x & 31;
  const int wave  = (int)((blockIdx.x * blockDim.x + threadIdx.x) >> 5);
  const int row0  = wave * 16;
  if (row0 >= nrows) return;

  const int l15   = lane & 15;
  const int khalf = (lane >> 4) * 2;    // 0 for lanes 0-15, 2 for lanes 16-31

  int arow = row0 + l15;
  if (arow >= nrows) arow = nrows - 1;  // clamp (N % 16 == 0 here, but safe)

  const float* __restrict__ ha  = h   + (size_t)arow * IN_DIM + khalf;
  const float* __restrict__ wb0 = Wfc + (size_t)(l15 +  0) * IN_DIM + khalf;
  const float* __restrict__ wb1 = Wfc + (size_t)(l15 + 16) * IN_DIM + khalf;
  const float* __restrict__ wb2 = Wfc + (size_t)(l15 + 32) * IN_DIM + khalf;
  const float* __restrict__ wb3 = Wfc + (size_t)(l15 + 48) * IN_DIM + khalf;

  v8f acc0 = {};
  v8f acc1 = {};
  v8f acc2 = {};
  v8f acc3 = {};

#pragma unroll 8
  for (int k = 0; k < IN_DIM; k += 4) {
    v2f a  = *(const v2f*)(ha  + k);
    v2f b0 = *(const v2f*)(wb0 + k);
    v2f b1 = *(const v2f*)(wb1 + k);
    v2f b2 = *(const v2f*)(wb2 + k);
    v2f b3 = *(const v2f*)(wb3 + k);
    acc0 = __builtin_amdgcn_wmma_f32_16x16x4_f32(false, a, false, b0,
                                                 (short)0, acc0, false, false);
    acc1 = __builtin_amdgcn_wmma_f32_16x16x4_f32(false, a, false, b1,
                                                 (short)0, acc1, false, false);
    acc2 = __builtin_amdgcn_wmma_f32_16x16x4_f32(false, a, false, b2,
                                                 (short)0, acc2, false, false);
    acc3 = __builtin_amdgcn_wmma_f32_16x16x4_f32(false, a, false, b3,
                                                 (short)0, acc3, false, false);
  }

  const float bias0 = bfc[l15 +  0];
  const float bias1 = bfc[l15 + 16];
  const float bias2 = bfc[l15 + 32];
  const float bias3 = bfc[l15 + 48];

  const int mtop = (lane >> 4) * 8;     // C/D: M = r + 8*(lane/16)
#pragma unroll
  for (int r = 0; r < 8; ++r) {
    const int m = row0 + mtop + r;
    if (m < nrows) {
      float* __restrict__ zr = z + (size_t)m * OUT_DIM + l15;
      zr[0]  = acc0[r] + bias0;
      zr[16] = acc1[r] + bias1;
      zr[32] = acc2[r] + bias2;
      zr[48] = acc3[r] + bias3;
    }
  }
}

// ---------------------------------------------------------------------------
// Edge phase: one wave32 per edge. Lane t owns channels {2t, 2t+1}.
// att = leaky_relu(z[src].w1 + z[dst].w2 + b_att); out[dst] += z[src]*att.
// Butterfly reduction across the 32 lanes; native f32 global atomics.
// ---------------------------------------------------------------------------
__global__ void gat_edge_kernel(const int* __restrict__ src,
                                const int* __restrict__ dst,
                                const float* __restrict__ z,
                                const float* __restrict__ watt,
                                const float* __restrict__ batt,
                                float* __restrict__ out,
                                int nedges) {
  const int lane = threadIdx.x & 31;
  const int e    = (int)((blockIdx.x * blockDim.x + threadIdx.x) >> 5);
  if (e >= nedges) return;

  const int s = src[e];
  const int d = dst[e];

  const v2f zs = *(const v2f*)(z + (size_t)s * OUT_DIM + 2 * lane);
  const v2f zd = *(const v2f*)(z + (size_t)d * OUT_DIM + 2 * lane);
  const v2f a1 = *(const v2f*)(watt + 2 * lane);            // w1
  const v2f a2 = *(const v2f*)(watt + OUT_DIM + 2 * lane);  // w2

  float p = zs.x * a1.x + zs.y * a1.y + zd.x * a2.x + zd.y * a2.y;
  // wave32 butterfly sum (width 32 — CDNA5 is wave32, never 64)
  p += __shfl_xor(p, 16, 32);
  p += __shfl_xor(p,  8, 32);
  p += __shfl_xor(p,  4, 32);
  p += __shfl_xor(p,  2, 32);
  p += __shfl_xor(p,  1, 32);
  p += batt[0];

  const float att = (p > 0.0f) ? p : 0.01f * p;   // leaky_relu, slope 0.01

  float* __restrict__ o = out + (size_t)d * OUT_DIM + 2 * lane;
  unsafeAtomicAdd(o,     zs.x * att);   // -> global_atomic_add_f32
  unsafeAtomicAdd(o + 1, zs.y * att);
}

// ---------------------------------------------------------------------------
// Host-side launch (graph-capture safe: kernels only, all on `stream`).
// Inputs: 0=adj(2,E) i32, 1=h(N,128) f32, 2=W_fc(64,128) f32, 3=b_fc(64),
//         4=W_att(1,128) f32, 5=b_att(1) f32.  Output: (N,64) f32.
// ---------------------------------------------------------------------------
extern "C" void kernel_launch(void* const* d_in, const int* in_sizes, int n_in,
                              void* d_out, int out_size, void* d_ws, size_t ws_size,
                              hipStream_t stream) {
  const int*   adj  = (const int*)d_in[0];
  const float* h    = (const float*)d_in[1];
  const float* Wfc  = (const float*)d_in[2];
  const float* bfc  = (const float*)d_in[3];
  const float* Watt = (const float*)d_in[4];
  const float* batt = (const float*)d_in[5];
  float*       out  = (float*)d_out;

  const int E = in_sizes[0] / 2;
  const int N = in_sizes[1] / IN_DIM;
  const int* src = adj;        // adj[0, :]
  const int* dst = adj + E;    // adj[1, :]

  float* z = (float*)d_ws;     // scratch: N * 64 floats (12.8 MB)

  // 1) zero output (poisoned by harness)
  const int n4 = (N * OUT_DIM) / 4;
  gat_zero_kernel<<<(n4 + 255) / 256, 256, 0, stream>>>((float4*)out, n4);

  // 2) z = h @ W_fc^T + b_fc  (WMMA f32; 16 rows per wave, 8 waves/block)
  const int tiles   = (N + 15) / 16;
  const int gblocks = (tiles + 7) / 8;
  gat_gemm_wmma_kernel<<<gblocks, 256, 0, stream>>>(h, Wfc, bfc, z, N);

  // 3) edge scatter (1 wave per edge, 8 edges per 256-thread block)
  const int eblocks = (E + 7) / 8;
  gat_edge_kernel<<<eblocks, 256, 0, stream>>>(src, dst, z, Watt, batt, out, E);
}